// M18BiasCompiler_32899449487392
// MI455X (gfx1250) — compile-verified
//
#include <hip/hip_runtime.h>

typedef float v2f __attribute__((ext_vector_type(2)));
typedef float v4f __attribute__((ext_vector_type(4)));
typedef float v8f __attribute__((ext_vector_type(8)));

#define B_ 4
#define K_ 64
#define R_ 16
#define H_ 16
#define S_ 2048
#define CHUNKS 16
#define ROWS_PER_CHUNK (S_ / CHUNKS)   // 128

// ---------------------------------------------------------------------------
// Kernel 1: anchor_salience[b,h,j] = sum_r (mean_i adj[b,i,j,r]) * W[r,h]
// One block per batch b, 128 threads (4 waves). Reduction into LDS, then each
// wave computes one 16(j)x16(h) tile with 4 chained V_WMMA_F32_16X16X4_F32.
// ---------------------------------------------------------------------------
__global__ __launch_bounds__(128) void salience_kernel(
    const float* __restrict__ adj,   // [B][K][K][R]
    const float* __restrict__ W,     // [R][H]
    float* __restrict__ sal)         // [B][H][K]
{
    __shared__ float lds_m[K_][R_];  // M[j][r] = mean_i adj[b,i,j,r]

    const int b = blockIdx.x;
    const int t = threadIdx.x;
    const float* adjb = adj + (size_t)b * K_ * K_ * R_;

    // i-reduction: entry e = j*R + r ; consecutive threads -> contiguous loads
    for (int e = t; e < K_ * R_; e += 128) {
        float acc = 0.0f;
        #pragma unroll 4
        for (int i = 0; i < K_; ++i)
            acc += adjb[(size_t)i * (K_ * R_) + e];
        lds_m[e / R_][e % R_] = acc * (1.0f / (float)K_);
    }
    __syncthreads();

    // WMMA phase: wave w handles j-tile [w*16, w*16+16)
    const int wave = t >> 5;
    const int lane = t & 31;
    const int m    = lane & 15;              // A row (j within tile) / B col (h)
    const int half = (lane >> 4) << 1;       // 0 for lanes 0-15, 2 for 16-31

    v8f c = {};
    #pragma unroll
    for (int kk = 0; kk < 4; ++kk) {
        const int kb = kk * 4 + half;        // K index for VGPR0; +1 for VGPR1
        v2f a, bm;
        // A: 16x4 fp32, ISA layout: V0 = K {0|2}, V1 = K {1|3} per lane half
        a.x = lds_m[wave * 16 + m][kb];
        a.y = lds_m[wave * 16 + m][kb + 1];
        // B: 4x16 fp32, rows striped across lanes within a VGPR
        bm.x = W[kb * H_ + m];
        bm.y = W[(kb + 1) * H_ + m];
        c = __builtin_amdgcn_wmma_f32_16x16x4_f32(
                /*neg_a=*/false, a, /*neg_b=*/false, bm,
                /*c_mod=*/(short)0, c, /*reuse_a=*/false, /*reuse_b=*/false);
    }

    // D layout: VGPR v -> row M = v (+8 for lanes 16-31), col N = lane%16
    #pragma unroll
    for (int v = 0; v < 8; ++v) {
        const int j = wave * 16 + v + ((lane >> 4) << 3);
        const int h = m;
        sal[((size_t)b * H_ + h) * K_ + j] = c[v];
    }
}

// ---------------------------------------------------------------------------
// Kernel 2: out[b,h,row,col] = (col == idx[b,k]) ? sal[b,h,k] : 0, all rows.
// Grid: (B*H, CHUNKS). Build the 8KB row template in LDS once, cache it in
// registers, stream it to 128 rows with coalesced non-temporal float4 stores.
// Entire 1 GiB output is written exactly once -> HBM-store-bound (~46 us).
// ---------------------------------------------------------------------------
__global__ __launch_bounds__(256) void scatter_fill_kernel(
    const float* __restrict__ sal,   // [B][H][K]
    const int*   __restrict__ idx,   // [B][K]
    float*       __restrict__ out)   // [B][H][S][S]
{
    __shared__ float row[S_];

    const int bh = blockIdx.x;
    const int b  = bh / H_;
    const int h  = bh % H_;
    const int t  = threadIdx.x;

    v4f* row4 = (v4f*)row;
    v4f  z    = {};
    row4[t]       = z;
    row4[t + 256] = z;
    __syncthreads();

    if (t == 0) {
        // sequential k order -> deterministic last-write-wins on duplicate idx
        for (int k = 0; k < K_; ++k) {
            int c = idx[b * K_ + k];
            if ((unsigned)c < (unsigned)S_)
                row[c] = sal[((size_t)b * H_ + h) * K_ + k];
        }
    }
    __syncthreads();

    const v4f va = row4[t];
    const v4f vb = row4[t + 256];

    float* outbh = out + (((size_t)bh) * S_ +
                          (size_t)blockIdx.y * ROWS_PER_CHUNK) * S_;
    for (int r = 0; r < ROWS_PER_CHUNK; ++r) {
        v4f* o = (v4f*)(outbh + (size_t)r * S_);
        __builtin_nontemporal_store(va, o + t);
        __builtin_nontemporal_store(vb, o + t + 256);
    }
}

// ---------------------------------------------------------------------------
extern "C" void kernel_launch(void* const* d_in, const int* in_sizes, int n_in,
                              void* d_out, int out_size, void* d_ws, size_t ws_size,
                              hipStream_t stream) {
    (void)in_sizes; (void)n_in; (void)out_size; (void)ws_size;

    const float* adj = (const float*)d_in[0];   // [B,K,K,R] fp32
    const float* W   = (const float*)d_in[1];   // [R,H]     fp32
    const int*   idx = (const int*)d_in[2];     // [B,K]     int32
    float*       out = (float*)d_out;           // [B,H,S,S] fp32
    float*       sal = (float*)d_ws;            // scratch: B*H*K floats (16 KB)

    salience_kernel<<<B_, 128, 0, stream>>>(adj, W, sal);
    scatter_fill_kernel<<<dim3(B_ * H_, CHUNKS), 256, 0, stream>>>(sal, idx, out);
}